// convMultiheadAttetionV2_9165460210060
// MI455X (gfx1250) — compile-verified
//
#include <hip/hip_runtime.h>
#include <hip/hip_bf16.h>
#include <stdint.h>

typedef __attribute__((ext_vector_type(2))) float v2f;
typedef __attribute__((ext_vector_type(8))) float v8f;

#define AS1 __attribute__((address_space(1)))
#define AS3 __attribute__((address_space(3)))

#define L_PIX 16384
#define HDIM  128
#define CH_STRIDE 208   // LDS floats per channel (3*66=198 used; pad so the
                        // two lane-halves' channels hit disjoint bank ranges)

// Async global->LDS copy of one dword, with inline-asm fallback.
__device__ __forceinline__ void async_g2l_b32(const float* gp, float* lp) {
#if __has_builtin(__builtin_amdgcn_global_load_async_to_lds_b32)
    __builtin_amdgcn_global_load_async_to_lds_b32(
        (AS1 int*)(uintptr_t)gp, (AS3 int*)(AS3 void*)lp, 0, 0);
#else
    uint32_t lo = (uint32_t)(uintptr_t)(AS3 float*)(AS3 void*)lp;
    uint64_t ga = (uint64_t)(uintptr_t)gp;
    asm volatile("global_load_async_to_lds_b32 %0, %1, off"
                 :: "v"(lo), "v"(ga) : "memory");
#endif
}

__device__ __forceinline__ void wait_asynccnt0() {
#if __has_builtin(__builtin_amdgcn_s_wait_asynccnt)
    __builtin_amdgcn_s_wait_asynccnt(0);
#else
    asm volatile("s_wait_asynccnt 0x0" ::: "memory");
#endif
}

// ---------------------------------------------------------------------------
// Kernel 0: reorder proj weights (O,C,3,3) -> Wt[o][tap*64 + c] so the WMMA
// A-fragment (2 consecutive K for a fixed outch) is a contiguous float2 load.
// ---------------------------------------------------------------------------
__global__ __launch_bounds__(256) void reorder_w_kernel(
    const float* __restrict__ pw, float* __restrict__ wt)
{
    int i = blockIdx.x * 256 + threadIdx.x;
    if (i >= 64 * 576) return;
    int o = i / 576;
    int K = i % 576;
    int tap = K >> 6;
    int c   = K & 63;
    wt[i] = pw[o * 576 + c * 9 + tap];
}

// ---------------------------------------------------------------------------
// Kernel 1: QKV 1x1 conv as GEMM (192 x 64) @ (64 x L) per batch.
// One wave: 16 outch x 64 px tile, 16 K-steps x 4 wmma (f32 16x16x4).
// ---------------------------------------------------------------------------
__global__ __launch_bounds__(256) void qkv_kernel(
    const float* __restrict__ x, const float* __restrict__ w,
    const float* __restrict__ bias,
    float* __restrict__ qp, float* __restrict__ kp, float* __restrict__ vp)
{
    const int L = L_PIX;
    int wave = blockIdx.x * 8 + (threadIdx.x >> 5);
    int lane = threadIdx.x & 31;
    int m = lane & 15;          // M (outch) / N (pixel) index within tile
    int g = lane >> 4;          // K half-select

    int b   = wave / (12 * 256);
    int rem = wave % (12 * 256);
    int oc0 = (rem / 256) * 16;
    int px0 = (rem % 256) * 64;

    const float* xb = x + (size_t)b * 64 * L;
    const float* wrow = w + (size_t)(oc0 + m) * 64;

    v8f acc[4] = {};
    for (int kb = 0; kb < 64; kb += 4) {
        int kk = kb + 2 * g;                    // this lane holds K = kk, kk+1
        v2f a;
        a.x = wrow[kk];
        a.y = wrow[kk + 1];
        const float* xr0 = xb + (size_t)kk * L + px0 + m;
        const float* xr1 = xr0 + L;
#pragma unroll
        for (int t = 0; t < 4; ++t) {
            v2f bv;
            bv.x = xr0[16 * t];
            bv.y = xr1[16 * t];
            acc[t] = __builtin_amdgcn_wmma_f32_16x16x4_f32(
                false, a, false, bv, (short)0, acc[t], false, false);
        }
    }

#pragma unroll
    for (int t = 0; t < 4; ++t) {
#pragma unroll
        for (int r = 0; r < 8; ++r) {
            int ch  = oc0 + r + 8 * g;
            int pix = px0 + 16 * t + m;
            float val = acc[t][r] + bias[ch];
            if (ch < 128) val = fmaxf(val, 0.0f);   // ReLU on q and k
            int plane = ch >> 6;                    // 0=q 1=k 2=v
            float* dst = (plane == 0) ? qp : (plane == 1) ? kp : vp;
            dst[(size_t)b * 64 * L + (size_t)(ch & 63) * L + pix] = val;
        }
    }
}

// ---------------------------------------------------------------------------
// Kernel 2: kv[b,h,d,j] = sum_l k_shift[d][l]*v[j][l],  ksum[b,h,d].
// One block per (b,h), 32 waves; wave-shuffle reduce; deterministic.
// ---------------------------------------------------------------------------
__global__ __launch_bounds__(1024) void kvred_kernel(
    const float* __restrict__ kp, const float* __restrict__ vp,
    float* __restrict__ kv, float* __restrict__ ksum)
{
    const int L = L_PIX;
    int bh = blockIdx.x;            // 0..31
    int b = bh >> 3, h = bh & 7;
    int wid = threadIdx.x >> 5, lane = threadIdx.x & 31;

    const float* kb8 = kp + ((size_t)b * 64 + h * 8) * L;
    const float* vb8 = vp + ((size_t)b * 64 + h * 8) * L;

    for (int e = wid; e < 648; e += 32) {
        int d   = (e < 576) ? (e >> 3) : (e - 576);
        int j   = e & 7;
        int c   = d / 9;
        int tap = d % 9;
        int dy = tap / 3 - 1, dx = tap % 3 - 1;
        const float* kr = kb8 + (size_t)c * L;
        const float* vr = vb8 + (size_t)j * L;
        bool want_kv = (e < 576);

        float sum = 0.0f;
        for (int p = lane; p < L; p += 32) {
            int y = p >> 7, xx = p & 127;
            int ys = y + dy, xs = xx + dx;
            float kval = ((unsigned)ys < (unsigned)HDIM &&
                          (unsigned)xs < (unsigned)HDIM)
                             ? kr[ys * HDIM + xs] : 0.0f;
            sum += want_kv ? (kval * vr[p]) : kval;
        }
        for (int s = 16; s > 0; s >>= 1) sum += __shfl_xor(sum, s, 32);
        if (lane == 0) {
            if (want_kv) kv[(size_t)bh * 576 + e] = sum;
            else         ksum[(size_t)bh * 72 + (e - 576)] = sum;
        }
    }
}

// ---------------------------------------------------------------------------
// Kernel 3: out[b,8h+j,l] = (sum_d q_shift[d][l]*kv[d][j]) /
//                           (sum_d q_shift[d][l]*ksum[d] + EPS)
// ---------------------------------------------------------------------------
__global__ __launch_bounds__(256) void attnout_kernel(
    const float* __restrict__ qp, const float* __restrict__ kv,
    const float* __restrict__ ksum, float* __restrict__ attn)
{
    const int L = L_PIX;
    __shared__ float skv[576];
    __shared__ float sks[72];

    int bh = blockIdx.x >> 6;
    int chunk = blockIdx.x & 63;
    int b = bh >> 3, h = bh & 7;
    int t = threadIdx.x;

    for (int i = t; i < 576; i += 256) skv[i] = kv[(size_t)bh * 576 + i];
    if (t < 72) sks[t] = ksum[(size_t)bh * 72 + t];
    __syncthreads();

    int p = chunk * 256 + t;
    int y = p >> 7, xx = p & 127;
    const float* qb8 = qp + ((size_t)b * 64 + h * 8) * L;

    float acc[8] = {0.f, 0.f, 0.f, 0.f, 0.f, 0.f, 0.f, 0.f};
    float den = 0.0f;
    for (int tap = 0; tap < 9; ++tap) {
        int dy = tap / 3 - 1, dx = tap % 3 - 1;
        int ys = y + dy, xs = xx + dx;
        bool valid = (unsigned)ys < (unsigned)HDIM &&
                     (unsigned)xs < (unsigned)HDIM;
        int sp = ys * HDIM + xs;
#pragma unroll
        for (int c = 0; c < 8; ++c) {
            float qv = valid ? qb8[(size_t)c * L + sp] : 0.0f;
            int d = c * 9 + tap;
            den += qv * sks[d];
            const float* kvr = &skv[d * 8];
#pragma unroll
            for (int j = 0; j < 8; ++j) acc[j] += qv * kvr[j];
        }
    }
    float inv = 1.0f / (den + 1e-7f);
    float* ob = attn + ((size_t)b * 64 + h * 8) * L;
#pragma unroll
    for (int j = 0; j < 8; ++j) ob[(size_t)j * L + p] = acc[j] * inv;
}

// ---------------------------------------------------------------------------
// Kernel 4: 3x3 proj conv as implicit GEMM (64 x 576) @ (576 x L) per batch.
// Block = 64 outch x 64 px strip. Activation halo (64ch x 3row x 66col,
// border-zeroed) staged in LDS via async global->LDS; WMMA B-fragments come
// from LDS with no masking in the MAC loop. 8 waves = 4 oc-tiles x 2 px-grps,
// 288 wmma per wave.
// ---------------------------------------------------------------------------
__global__ __launch_bounds__(256) void proj_kernel(
    const float* __restrict__ attn, const float* __restrict__ wt,
    const float* __restrict__ pb, float* __restrict__ out)
{
    const int L = L_PIX;
    __shared__ float smem[64 * CH_STRIDE];   // 53,248 bytes

    int tid  = threadIdx.x;
    int wid  = tid >> 5;
    int lane = tid & 31;
    int m = lane & 15, g = lane >> 4;

    int b     = blockIdx.x >> 8;          // 4 batches
    int strip = blockIdx.x & 255;         // 256 strips of 64 px
    int y0 = strip >> 1;
    int x0 = (strip & 1) * 64;

    const float* ab = attn + (size_t)b * 64 * L;

    // ---- stage 64ch x 3row x 66col halo into LDS (async global->LDS) ----
    for (int idx = tid; idx < 64 * 198; idx += 256) {
        int c  = idx / 198;
        int r  = idx % 198;
        int ry = r / 66;          // 0..2
        int cx = r % 66;          // 0..65
        int ys = y0 + ry - 1;
        int xs = x0 + cx - 1;
        float* lp = &smem[c * CH_STRIDE + ry * 66 + cx];
        if ((unsigned)ys < (unsigned)HDIM && (unsigned)xs < (unsigned)HDIM) {
            async_g2l_b32(ab + (size_t)c * L + ys * HDIM + xs, lp);
        } else {
            *lp = 0.0f;           // materialize border zeros once
        }
    }
    wait_asynccnt0();
    __syncthreads();

    // ---- MAC loop: A from L2-resident reordered weights, B from LDS ----
    int oct = wid & 3;                    // oc tile 0..3
    int pg  = wid >> 2;                   // pixel group 0..1 (32 px each)
    int oc0 = oct * 16;
    const float* wbase = wt + (size_t)(oc0 + m) * 576;

    v8f acc[2] = {};
    for (int tap = 0; tap < 9; ++tap) {
        int dy = tap / 3 - 1, dx = tap % 3 - 1;
        int pxoff = (dy + 1) * 66 + (dx + 1) + pg * 32 + m;
        const float* wrow = wbase + tap * 64;
        for (int cb = 0; cb < 64; cb += 4) {
            int kk = cb + 2 * g;          // input-channel pair for this lane
            v2f a;
            a.x = wrow[kk];
            a.y = wrow[kk + 1];
            int lo = kk * CH_STRIDE + pxoff;
#pragma unroll
            for (int t = 0; t < 2; ++t) {
                v2f bv;
                bv.x = smem[lo + 16 * t];
                bv.y = smem[lo + 16 * t + CH_STRIDE];
                acc[t] = __builtin_amdgcn_wmma_f32_16x16x4_f32(
                    false, a, false, bv, (short)0, acc[t], false, false);
            }
        }
    }

#pragma unroll
    for (int t = 0; t < 2; ++t) {
#pragma unroll
        for (int r = 0; r < 8; ++r) {
            int ch  = oc0 + r + 8 * g;
            int pix = y0 * HDIM + x0 + pg * 32 + 16 * t + m;
            out[(size_t)b * 64 * L + (size_t)ch * L + pix] = acc[t][r] + pb[ch];
        }
    }
}

// ---------------------------------------------------------------------------
extern "C" void kernel_launch(void* const* d_in, const int* in_sizes, int n_in,
                              void* d_out, int out_size, void* d_ws, size_t ws_size,
                              hipStream_t stream) {
    (void)in_sizes; (void)n_in; (void)out_size; (void)ws_size;
    const float* x    = (const float*)d_in[0];  // (4,64,128,128)
    const float* qkvw = (const float*)d_in[1];  // (192,64)
    const float* qkvb = (const float*)d_in[2];  // (192,)
    const float* pw   = (const float*)d_in[3];  // (64,64,3,3)
    const float* pb   = (const float*)d_in[4];  // (64,)
    float* out = (float*)d_out;                 // (4,64,128,128)

    float* ws = (float*)d_ws;
    const size_t PLANE = (size_t)4 * 64 * L_PIX;   // 4,194,304 floats
    float* wsq    = ws;
    float* wsk    = ws + PLANE;
    float* wsv    = ws + 2 * PLANE;
    float* wsattn = ws + 3 * PLANE;
    float* wskv   = ws + 4 * PLANE;            // 32*576 floats
    float* wsks   = wskv + 32 * 576;           // 32*72 floats
    float* wswt   = wsks + 32 * 72;            // 64*576 floats

    reorder_w_kernel<<<(64 * 576 + 255) / 256, 256, 0, stream>>>(pw, wswt);
    qkv_kernel<<<1536, 256, 0, stream>>>(x, qkvw, qkvb, wsq, wsk, wsv);
    kvred_kernel<<<32, 1024, 0, stream>>>(wsk, wsv, wskv, wsks);
    attnout_kernel<<<2048, 256, 0, stream>>>(wsq, wskv, wsks, wsattn);
    proj_kernel<<<1024, 256, 0, stream>>>(wsattn, wswt, pb, out);
}